// CandidateVoteModule_26688926777516
// MI455X (gfx1250) — compile-verified
//
#include <hip/hip_runtime.h>
#include <hip/hip_bf16.h>
#include <math.h>

#define BB 4
#define NN 8192
#define IN_DIM 32
#define OUT_DIM 128
#define NPOINT 1024
#define NSAMPLE 32
#define RADIUS 0.2f
#define BN_EPS 1e-5f

#define PT_SA (BB * NPOINT * NSAMPLE)   // 131072 columns for SA gemms
#define PT_FP (BB * NN)                 // 32768 columns for FP / vote gemms
#define C_SA0 36                        // 35 real channels + 1 zero pad (K mult of 4)
#define M_C3P 144                       // 132 vote-head rows padded to 9 m-tiles

typedef float v2f __attribute__((ext_vector_type(2)));
typedef float v8f __attribute__((ext_vector_type(8)));

// ---------------------------------------------------------------------------
// 1) Farthest point sampling: one block per batch, 512 threads * 16 pts each.
//    Wave32 shuffle reduction -> only 2 barriers per serial iteration.
// ---------------------------------------------------------------------------
__global__ void fps_kernel(const float* __restrict__ xyz,
                           float* __restrict__ new_xyz) {
  const int b = blockIdx.x;
  const int t = threadIdx.x;          // 0..511
  const int lane = t & 31;
  const int wv = t >> 5;              // 0..15
  const int PPT = NN / 512;           // 16 points per thread
  const float* px = xyz + (size_t)b * NN * 3;

  float dx[16], dy[16], dz[16], dist[16];
#pragma unroll
  for (int i = 0; i < PPT; ++i) {
    int p = t * PPT + i;
    dx[i] = px[p * 3 + 0];
    dy[i] = px[p * 3 + 1];
    dz[i] = px[p * 3 + 2];
    dist[i] = 1e10f;
  }

  __shared__ float s_val[16];
  __shared__ int   s_idx[16];
  __shared__ int   s_far;

  int far = 0;
  for (int it = 0; it < NPOINT; ++it) {
    // uniform broadcast load of current centroid (L2-resident)
    const float cx = px[far * 3 + 0];
    const float cy = px[far * 3 + 1];
    const float cz = px[far * 3 + 2];
    if (t == 0) {
      float* o = new_xyz + ((size_t)b * NPOINT + it) * 3;
      o[0] = cx; o[1] = cy; o[2] = cz;   // emit incoming 'far' (matches scan)
    }

    float best = -1.0f; int bi = t * PPT;
#pragma unroll
    for (int i = 0; i < PPT; ++i) {
      float ax = dx[i] - cx, ay = dy[i] - cy, az = dz[i] - cz;
      float d = ax * ax + ay * ay + az * az;
      d = fminf(dist[i], d);
      dist[i] = d;
      if (d > best) { best = d; bi = t * PPT + i; }
    }
    // wave32 argmax (tie -> smaller index, matching argmax-first semantics)
#pragma unroll
    for (int off = 16; off > 0; off >>= 1) {
      float ov = __shfl_down(best, off);
      int   oi = __shfl_down(bi, off);
      if (ov > best || (ov == best && oi < bi)) { best = ov; bi = oi; }
    }
    if (lane == 0) { s_val[wv] = best; s_idx[wv] = bi; }
    __syncthreads();
    if (wv == 0) {
      float v = (lane < 16) ? s_val[lane] : -2.0f;
      int idx = (lane < 16) ? s_idx[lane] : 0;
#pragma unroll
      for (int off = 8; off > 0; off >>= 1) {
        float ov = __shfl_down(v, off);
        int   oi = __shfl_down(idx, off);
        if (ov > v || (ov == v && oi < idx)) { v = ov; idx = oi; }
      }
      if (lane == 0) s_far = idx;
    }
    __syncthreads();
    far = s_far;
  }
}

// ---------------------------------------------------------------------------
// 2) Ball query: one wave32 per center; ballot+popc keeps first-32 ascending
//    index semantics of the reference's top_k trick.
// ---------------------------------------------------------------------------
__global__ void ball_query_kernel(const float* __restrict__ xyz,
                                  const float* __restrict__ new_xyz,
                                  int* __restrict__ gidx) {
  const int wid = (blockIdx.x * blockDim.x + threadIdx.x) >> 5;
  const int lane = threadIdx.x & 31;
  if (wid >= BB * NPOINT) return;
  const int b = wid / NPOINT;
  const float cx = new_xyz[wid * 3 + 0];
  const float cy = new_xyz[wid * 3 + 1];
  const float cz = new_xyz[wid * 3 + 2];
  const float* px = xyz + (size_t)b * NN * 3;
  int* out = gidx + (size_t)wid * NSAMPLE;

  const float r2 = RADIUS * RADIUS;
  int found = 0;
  int first = -1;
  for (int base = 0; base < NN && found < NSAMPLE; base += 32) {
    const int p = base + lane;
    float ax = px[p * 3 + 0] - cx;
    float ay = px[p * 3 + 1] - cy;
    float az = px[p * 3 + 2] - cz;
    bool in = (ax * ax + ay * ay + az * az) < r2;
    unsigned mask = (unsigned)__ballot(in);
    if (in) {
      int rank = found + __popc(mask & ((1u << lane) - 1u));
      if (rank < NSAMPLE) out[rank] = p;
    }
    if (first < 0 && mask) first = base + (__ffs(mask) - 1);
    found += __popc(mask);
  }
  if (first < 0) first = 0;
  found = found < NSAMPLE ? found : NSAMPLE;
  for (int s = found + lane; s < NSAMPLE; s += 32) out[s] = first;
}

// ---------------------------------------------------------------------------
// 3) Build grouped SA input: C_SA0=36 rows (row 35 zero pad) x PT_SA cols
// ---------------------------------------------------------------------------
__global__ void group_build_kernel(const float* __restrict__ xyz,
                                   const float* __restrict__ feats,
                                   const float* __restrict__ new_xyz,
                                   const int* __restrict__ gidx,
                                   float* __restrict__ H) {
  const int p = blockIdx.x * blockDim.x + threadIdx.x;
  if (p >= PT_SA) return;
  const int pc = p >> 5;               // (b,pt) center id
  const int b = pc >> 10;              // NPOINT = 1024
  const int g = gidx[p];
  const float* src = xyz + ((size_t)b * NN + g) * 3;
  const float* ctr = new_xyz + (size_t)pc * 3;
  const float inv_r = 1.0f / RADIUS;
  H[0 * (size_t)PT_SA + p] = (src[0] - ctr[0]) * inv_r;
  H[1 * (size_t)PT_SA + p] = (src[1] - ctr[1]) * inv_r;
  H[2 * (size_t)PT_SA + p] = (src[2] - ctr[2]) * inv_r;
#pragma unroll
  for (int c = 0; c < IN_DIM; ++c) {
    H[(size_t)(3 + c) * PT_SA + p] = feats[((size_t)b * IN_DIM + c) * NN + g];
  }
  H[(size_t)35 * PT_SA + p] = 0.0f;    // zero pad channel -> K=36
}

// ---------------------------------------------------------------------------
// 4) Weight/bias padding: copy (M x K) into zero-filled (Mp x Kp)
// ---------------------------------------------------------------------------
__global__ void pad_kernel(const float* __restrict__ W, float* __restrict__ Wp,
                           int M, int K, int Mp, int Kp) {
  const int i = blockIdx.x * blockDim.x + threadIdx.x;
  if (i >= Mp * Kp) return;
  const int m = i / Kp, k = i % Kp;
  Wp[i] = (m < M && k < K) ? W[m * K + k] : 0.0f;
}

// ---------------------------------------------------------------------------
// 5) FP32 WMMA GEMM: Y[Mp x P] = W[Mp x K] * X[K x P] (+ bias).
//    K compile-time (multiple of 4), no guards, one-step software pipeline:
//    fragments for step k+4 are loaded before the 4 WMMAs of step k issue,
//    so the serial acc-chain hides L2 latency. EXEC all ones throughout.
// ---------------------------------------------------------------------------
#define GEMM_NT 4

template <int K, bool HAS_BIAS>
__global__ void gemm_wmma_kernel(const float* __restrict__ W,
                                 const float* __restrict__ X,
                                 float* __restrict__ Y,
                                 const float* __restrict__ bias,
                                 int P) {
  const int lane = threadIdx.x & 31;
  const int mtile = threadIdx.x >> 5;     // wave id within block = M tile
  const int half = lane >> 4;             // 0: K pair {k0,k0+1}, 1: {k0+2,k0+3}
  const int l = lane & 15;
  const int row0 = mtile * 16;
  const int col0 = blockIdx.x * (16 * GEMM_NT);
  const float* wp = W + (size_t)(row0 + l) * K + half * 2;   // 8B aligned
  const float* xp = X + (size_t)(half * 2) * P + col0 + l;

  v8f acc[GEMM_NT];
#pragma unroll
  for (int t = 0; t < GEMM_NT; ++t) acc[t] = (v8f){};

  // prologue: fragments for k0 = 0
  v2f a_cur = *(const v2f*)wp;
  v2f b_cur[GEMM_NT];
#pragma unroll
  for (int t = 0; t < GEMM_NT; ++t) {
    b_cur[t].x = xp[t * 16];
    b_cur[t].y = xp[(size_t)P + t * 16];
  }

#pragma unroll
  for (int k0 = 0; k0 < K; k0 += 4) {
    v2f a_nxt = {};
    v2f b_nxt[GEMM_NT];
#pragma unroll
    for (int t = 0; t < GEMM_NT; ++t) b_nxt[t] = (v2f){};
    if (k0 + 4 < K) {                   // compile-time resolved
      a_nxt = *(const v2f*)(wp + k0 + 4);
      const float* xn = xp + (size_t)(k0 + 4) * P;
#pragma unroll
      for (int t = 0; t < GEMM_NT; ++t) {
        b_nxt[t].x = xn[t * 16];
        b_nxt[t].y = xn[(size_t)P + t * 16];
      }
    }
#pragma unroll
    for (int t = 0; t < GEMM_NT; ++t) {
      acc[t] = __builtin_amdgcn_wmma_f32_16x16x4_f32(
          /*neg_a=*/false, a_cur, /*neg_b=*/false, b_cur[t],
          /*c_mod=*/(short)0, acc[t], /*reuse_a=*/false, /*reuse_b=*/false);
    }
    a_cur = a_nxt;
#pragma unroll
    for (int t = 0; t < GEMM_NT; ++t) b_cur[t] = b_nxt[t];
  }

#pragma unroll
  for (int r = 0; r < 8; ++r) {
    const int row = row0 + r + half * 8;
    float bv = 0.0f;
    if (HAS_BIAS) bv = bias[row];
#pragma unroll
    for (int t = 0; t < GEMM_NT; ++t) {
      Y[(size_t)row * P + col0 + t * 16 + l] = acc[t][r] + bv;
    }
  }
}

// ---------------------------------------------------------------------------
// 6) BatchNorm statistics -> fused scale = g/sqrt(var+eps), shift = b - mean*sc
// ---------------------------------------------------------------------------
__global__ void bn_stats_kernel(const float* __restrict__ X,
                                const float* __restrict__ gamma,
                                const float* __restrict__ beta,
                                float* __restrict__ ss, int C, int P) {
  const int c = blockIdx.x;
  const float* row = X + (size_t)c * P;
  float s = 0.0f, s2 = 0.0f;
  for (int i = threadIdx.x; i < P; i += blockDim.x) {
    float v = row[i];
    s += v; s2 += v * v;
  }
  __shared__ float sh[256], sh2[256];
  sh[threadIdx.x] = s; sh2[threadIdx.x] = s2;
  __syncthreads();
  for (int k = 128; k > 0; k >>= 1) {
    if (threadIdx.x < k) {
      sh[threadIdx.x] += sh[threadIdx.x + k];
      sh2[threadIdx.x] += sh2[threadIdx.x + k];
    }
    __syncthreads();
  }
  if (threadIdx.x == 0) {
    float mean = sh[0] / (float)P;
    float var = sh2[0] / (float)P - mean * mean;
    float sc = gamma[c] * rsqrtf(var + BN_EPS);
    ss[c] = sc;
    ss[c + C] = beta[c] - mean * sc;
  }
}

__global__ void bn_relu_kernel(float* __restrict__ X,
                               const float* __restrict__ ss, int C, int P) {
  const size_t i = (size_t)blockIdx.x * blockDim.x + threadIdx.x;
  if (i >= (size_t)C * P) return;
  const int c = (int)(i / P);
  float v = X[i] * ss[c] + ss[c + C];
  X[i] = v > 0.0f ? v : 0.0f;
}

// ---------------------------------------------------------------------------
// 7) Max over nsample: feat_sa[b,c,pt] = max_s H[c, (b,pt,s)]
// ---------------------------------------------------------------------------
__global__ void maxpool_kernel(const float* __restrict__ H,
                               float* __restrict__ feat_sa) {
  const int i = blockIdx.x * blockDim.x + threadIdx.x;
  if (i >= BB * OUT_DIM * NPOINT) return;
  const int pt = i % NPOINT;
  const int c = (i / NPOINT) % OUT_DIM;
  const int b = i / (NPOINT * OUT_DIM);
  const float* src = H + (size_t)c * PT_SA + ((size_t)b * NPOINT + pt) * NSAMPLE;
  float m = src[0];
#pragma unroll
  for (int s = 1; s < NSAMPLE; ++s) m = fmaxf(m, src[s]);
  feat_sa[i] = m;
}

// ---------------------------------------------------------------------------
// 8) Three-NN: per point find 3 nearest centers + interpolation weights
// ---------------------------------------------------------------------------
__global__ void three_nn_kernel(const float* __restrict__ xyz,
                                const float* __restrict__ new_xyz,
                                int* __restrict__ nni,
                                float* __restrict__ nnw) {
  const int b = blockIdx.y;
  const int n = blockIdx.x * blockDim.x + threadIdx.x;
  __shared__ float sc[NPOINT * 3];
  for (int i = threadIdx.x; i < NPOINT * 3; i += blockDim.x)
    sc[i] = new_xyz[(size_t)b * NPOINT * 3 + i];
  __syncthreads();
  const float* p = xyz + ((size_t)b * NN + n) * 3;
  const float px = p[0], py = p[1], pz = p[2];
  float d0 = 1e30f, d1 = 1e30f, d2 = 1e30f;
  int i0 = 0, i1 = 0, i2 = 0;
  for (int j = 0; j < NPOINT; ++j) {
    float ax = px - sc[j * 3 + 0];
    float ay = py - sc[j * 3 + 1];
    float az = pz - sc[j * 3 + 2];
    float d = ax * ax + ay * ay + az * az;
    if (d < d0)      { d2 = d1; i2 = i1; d1 = d0; i1 = i0; d0 = d; i0 = j; }
    else if (d < d1) { d2 = d1; i2 = i1; d1 = d; i1 = j; }
    else if (d < d2) { d2 = d; i2 = j; }
  }
  const float r0 = 1.0f / (fmaxf(d0, 0.0f) + 1e-8f);
  const float r1 = 1.0f / (fmaxf(d1, 0.0f) + 1e-8f);
  const float r2 = 1.0f / (fmaxf(d2, 0.0f) + 1e-8f);
  const float inv = 1.0f / (r0 + r1 + r2);
  const size_t o = ((size_t)b * NN + n) * 3;
  nni[o + 0] = i0; nni[o + 1] = i1; nni[o + 2] = i2;
  nnw[o + 0] = r0 * inv; nnw[o + 1] = r1 * inv; nnw[o + 2] = r2 * inv;
}

// ---------------------------------------------------------------------------
// 9) Build FP input: rows 0..127 = 3-NN interp, rows 128..159 = features
// ---------------------------------------------------------------------------
__global__ void fp_build_kernel(const float* __restrict__ feats,
                                const float* __restrict__ feat_sa,
                                const int* __restrict__ nni,
                                const float* __restrict__ nnw,
                                float* __restrict__ FP) {
  const size_t i = (size_t)blockIdx.x * blockDim.x + threadIdx.x;
  if (i >= (size_t)(OUT_DIM + IN_DIM) * PT_FP) return;
  const int p = (int)(i % PT_FP);
  const int c = (int)(i / PT_FP);
  const int b = p / NN;
  const int n = p % NN;
  float v;
  if (c < OUT_DIM) {
    const int* ii = nni + (size_t)p * 3;
    const float* ww = nnw + (size_t)p * 3;
    const float* fs = feat_sa + ((size_t)b * OUT_DIM + c) * NPOINT;
    v = fs[ii[0]] * ww[0] + fs[ii[1]] * ww[1] + fs[ii[2]] * ww[2];
  } else {
    v = feats[((size_t)b * IN_DIM + (c - OUT_DIM)) * NN + n];
  }
  FP[(size_t)c * PT_FP + p] = v;
}

// ---------------------------------------------------------------------------
// 10) Final assembly: offsets, vote_xyz, sigmoid score, normalized vote feats
// ---------------------------------------------------------------------------
__global__ void final_kernel(const float* __restrict__ xyz,
                             const float* __restrict__ seed,
                             const float* __restrict__ net,
                             float* __restrict__ out) {
  const int p = blockIdx.x * blockDim.x + threadIdx.x;
  if (p >= PT_FP) return;
  const float ox = net[0 * (size_t)PT_FP + p];
  const float oy = net[1 * (size_t)PT_FP + p];
  const float oz = net[2 * (size_t)PT_FP + p];
  const float sg = net[3 * (size_t)PT_FP + p];

  float* vote_xyz = out;
  float* offset = out + (size_t)BB * NN * 3;
  float* scores = out + (size_t)2 * BB * NN * 3;
  float* vfeat = scores + (size_t)BB * NN;

  vote_xyz[(size_t)p * 3 + 0] = xyz[(size_t)p * 3 + 0] + ox;
  vote_xyz[(size_t)p * 3 + 1] = xyz[(size_t)p * 3 + 1] + oy;
  vote_xyz[(size_t)p * 3 + 2] = xyz[(size_t)p * 3 + 2] + oz;
  offset[(size_t)p * 3 + 0] = ox;
  offset[(size_t)p * 3 + 1] = oy;
  offset[(size_t)p * 3 + 2] = oz;
  scores[p] = 1.0f / (1.0f + expf(-sg));

  float nrm = 0.0f;
  for (int c = 0; c < OUT_DIM; ++c) {
    float v = seed[(size_t)c * PT_FP + p] + net[(size_t)(4 + c) * PT_FP + p];
    nrm += v * v;
  }
  const float inv = 1.0f / fmaxf(sqrtf(nrm), 1e-12f);
  for (int c = 0; c < OUT_DIM; ++c) {
    float v = seed[(size_t)c * PT_FP + p] + net[(size_t)(4 + c) * PT_FP + p];
    vfeat[(size_t)p * OUT_DIM + c] = v * inv;
  }
}

// ---------------------------------------------------------------------------
// Host launcher
// ---------------------------------------------------------------------------
extern "C" void kernel_launch(void* const* d_in, const int* in_sizes, int n_in,
                              void* d_out, int out_size, void* d_ws, size_t ws_size,
                              hipStream_t stream) {
  (void)in_sizes; (void)n_in; (void)out_size; (void)ws_size;
  const float* xyz   = (const float*)d_in[0];
  const float* feats = (const float*)d_in[1];
  const float* W_sa0 = (const float*)d_in[2];
  const float* g_sa0 = (const float*)d_in[3];
  const float* b_sa0 = (const float*)d_in[4];
  const float* W_sa1 = (const float*)d_in[5];
  const float* g_sa1 = (const float*)d_in[6];
  const float* b_sa1 = (const float*)d_in[7];
  const float* W_sa2 = (const float*)d_in[8];
  const float* g_sa2 = (const float*)d_in[9];
  const float* b_sa2 = (const float*)d_in[10];
  const float* W_fp0 = (const float*)d_in[11];
  const float* g_fp0 = (const float*)d_in[12];
  const float* b_fp0 = (const float*)d_in[13];
  const float* W_fp1 = (const float*)d_in[14];
  const float* g_fp1 = (const float*)d_in[15];
  const float* b_fp1 = (const float*)d_in[16];
  const float* Wc1   = (const float*)d_in[17];
  const float* g1    = (const float*)d_in[19];
  const float* b1    = (const float*)d_in[20];
  const float* Wc2   = (const float*)d_in[21];
  const float* g2    = (const float*)d_in[23];
  const float* b2    = (const float*)d_in[24];
  const float* Wc3   = (const float*)d_in[25];
  const float* bc3   = (const float*)d_in[26];
  // bc1/bc2 (d_in[18]/d_in[22]) cancel exactly under the following BatchNorm.

  // ---- workspace carve (256B aligned) ----
  size_t o = 0;
  auto carve = [&](size_t nbytes) -> char* {
    char* p = (char*)d_ws + o;
    o += (nbytes + 255) & ~(size_t)255;
    return p;
  };
  float* newXyz = (float*)carve((size_t)BB * NPOINT * 3 * 4);
  int*   gIdx   = (int*)  carve((size_t)BB * NPOINT * NSAMPLE * 4);
  float* ss     = (float*)carve((size_t)2 * 160 * 4);
  int*   nni    = (int*)  carve((size_t)BB * NN * 3 * 4);
  float* nnw    = (float*)carve((size_t)BB * NN * 3 * 4);
  float* featSa = (float*)carve((size_t)BB * OUT_DIM * NPOINT * 4);
  float* WSA0p  = (float*)carve((size_t)128 * C_SA0 * 4);    // padded W_sa0
  float* WC3p   = (float*)carve((size_t)M_C3P * 128 * 4);    // padded Wc3
  float* BC3p   = (float*)carve((size_t)M_C3P * 4);          // padded bc3
  float* H0     = (float*)carve((size_t)128 * PT_SA * 4);    // 64 MB ping
  float* H1     = (float*)carve((size_t)128 * PT_SA * 4);    // 64 MB pong
  // Reuse carved regions for FP / vote stages (liveness checked):
  float* FP0IN = H0;                                         // 160 x PT_FP (20 MB)
  float* FP0   = H1;                                         // 128 x PT_FP (16 MB)
  float* SEED  = (float*)((char*)H0 + (size_t)24 * 1024 * 1024); // 16 MB
  float* NET1  = (float*)((char*)H1 + (size_t)24 * 1024 * 1024); // 16 MB
  float* NET2  = H1;                                         // 16 MB (FP0 dead)
  float* NET3  = H0;                                         // 144 x PT_FP (18.9 MB < 24 MB, FP0IN dead)

  // ---- stage 0: weight padding ----
  pad_kernel<<<(128 * C_SA0 + 255) / 256, 256, 0, stream>>>(W_sa0, WSA0p, 128, 35, 128, C_SA0);
  pad_kernel<<<(M_C3P * 128 + 255) / 256, 256, 0, stream>>>(Wc3, WC3p, 132, 128, M_C3P, 128);
  pad_kernel<<<1, 256, 0, stream>>>(bc3, BC3p, 132, 1, M_C3P, 1);

  // ---- stage 1: FPS ----
  fps_kernel<<<BB, 512, 0, stream>>>(xyz, newXyz);

  // ---- stage 2: ball query + group build ----
  ball_query_kernel<<<(BB * NPOINT * 32) / 256, 256, 0, stream>>>(xyz, newXyz, gIdx);
  group_build_kernel<<<PT_SA / 256, 256, 0, stream>>>(xyz, feats, newXyz, gIdx, H0);

  // ---- stage 3: SA MLP (gemm -> bn stats -> bn+relu) x3 ----
  gemm_wmma_kernel<C_SA0, false><<<PT_SA / (16 * GEMM_NT), 256, 0, stream>>>(WSA0p, H0, H1, nullptr, PT_SA);
  bn_stats_kernel<<<128, 256, 0, stream>>>(H1, g_sa0, b_sa0, ss, 128, PT_SA);
  bn_relu_kernel<<<(128 * PT_SA) / 256, 256, 0, stream>>>(H1, ss, 128, PT_SA);

  gemm_wmma_kernel<128, false><<<PT_SA / (16 * GEMM_NT), 256, 0, stream>>>(W_sa1, H1, H0, nullptr, PT_SA);
  bn_stats_kernel<<<128, 256, 0, stream>>>(H0, g_sa1, b_sa1, ss, 128, PT_SA);
  bn_relu_kernel<<<(128 * PT_SA) / 256, 256, 0, stream>>>(H0, ss, 128, PT_SA);

  gemm_wmma_kernel<128, false><<<PT_SA / (16 * GEMM_NT), 256, 0, stream>>>(W_sa2, H0, H1, nullptr, PT_SA);
  bn_stats_kernel<<<128, 256, 0, stream>>>(H1, g_sa2, b_sa2, ss, 128, PT_SA);
  bn_relu_kernel<<<(128 * PT_SA) / 256, 256, 0, stream>>>(H1, ss, 128, PT_SA);

  // ---- stage 4: max pool over nsample ----
  maxpool_kernel<<<(BB * OUT_DIM * NPOINT) / 256, 256, 0, stream>>>(H1, featSa);

  // ---- stage 5: 3-NN interpolation + FP input build ----
  three_nn_kernel<<<dim3(NN / 256, BB), 256, 0, stream>>>(xyz, newXyz, nni, nnw);
  fp_build_kernel<<<((OUT_DIM + IN_DIM) * PT_FP) / 256, 256, 0, stream>>>(
      feats, featSa, nni, nnw, FP0IN);

  // ---- stage 6: FP MLPs ----
  gemm_wmma_kernel<160, false><<<PT_FP / (16 * GEMM_NT), 256, 0, stream>>>(W_fp0, FP0IN, FP0, nullptr, PT_FP);
  bn_stats_kernel<<<128, 256, 0, stream>>>(FP0, g_fp0, b_fp0, ss, 128, PT_FP);
  bn_relu_kernel<<<(128 * PT_FP) / 256, 256, 0, stream>>>(FP0, ss, 128, PT_FP);

  gemm_wmma_kernel<128, false><<<PT_FP / (16 * GEMM_NT), 256, 0, stream>>>(W_fp1, FP0, SEED, nullptr, PT_FP);
  bn_stats_kernel<<<128, 256, 0, stream>>>(SEED, g_fp1, b_fp1, ss, 128, PT_FP);
  bn_relu_kernel<<<(128 * PT_FP) / 256, 256, 0, stream>>>(SEED, ss, 128, PT_FP);

  // ---- stage 7: vote head ----
  gemm_wmma_kernel<128, false><<<PT_FP / (16 * GEMM_NT), 256, 0, stream>>>(Wc1, SEED, NET1, nullptr, PT_FP);
  bn_stats_kernel<<<128, 256, 0, stream>>>(NET1, g1, b1, ss, 128, PT_FP);
  bn_relu_kernel<<<(128 * PT_FP) / 256, 256, 0, stream>>>(NET1, ss, 128, PT_FP);

  gemm_wmma_kernel<128, false><<<PT_FP / (16 * GEMM_NT), 256, 0, stream>>>(Wc2, NET1, NET2, nullptr, PT_FP);
  bn_stats_kernel<<<128, 256, 0, stream>>>(NET2, g2, b2, ss, 128, PT_FP);
  bn_relu_kernel<<<(128 * PT_FP) / 256, 256, 0, stream>>>(NET2, ss, 128, PT_FP);

  gemm_wmma_kernel<128, true><<<PT_FP / (16 * GEMM_NT), 9 * 32, 0, stream>>>(WC3p, NET2, NET3, BC3p, PT_FP);

  // ---- stage 8: outputs ----
  final_kernel<<<PT_FP / 256, 256, 0, stream>>>(xyz, SEED, NET3, (float*)d_out);
}